// MultiPlaneReconstruction_33303176413566
// MI455X (gfx1250) — compile-verified
//
#include <hip/hip_runtime.h>

typedef float v2f __attribute__((ext_vector_type(2)));
typedef float v4f __attribute__((ext_vector_type(4)));
typedef float v8f __attribute__((ext_vector_type(8)));

#define S 80
#define SS 6400
#define SSS 512000L
#define CCH 128
#define B_ 2

// workspace layout (float elements)
#define CONV_SZ   (3L * B_ * CCH * SS)   // 4,915,200
#define FOUT_OFF  CONV_SZ
#define STAT1_OFF (2L * CONV_SZ)
#define STAT_STRIDE 164                  // [0]=S [1]=Q [2..81]=Row [82..161]=Col
#define STAT2_OFF (STAT1_OFF + 3L * B_ * CCH * STAT_STRIDE)

// ---------------------------------------------------------------------------
// Kernel 1: three 3x3 SAME convs on center slices, implicit GEMM via
// V_WMMA_F32_16X16X4_F32. One wave = 16 out-channels x 16 spatial positions.
// Zero padding is branchless: clamped (always-valid) address + v_cndmask on
// the loaded value, so EXEC stays all-ones and loads stay in clauses.
// ---------------------------------------------------------------------------
__global__ __launch_bounds__(256)
void conv_planes(const float* __restrict__ xin,
                 const float* __restrict__ w_ax,  const float* __restrict__ b_ax,
                 const float* __restrict__ w_cor, const float* __restrict__ b_cor,
                 const float* __restrict__ w_sag, const float* __restrict__ b_sag,
                 float* __restrict__ conv_out) {
  int wid  = blockIdx.x * 8 + (threadIdx.x >> 5);
  int lane = threadIdx.x & 31;
  int ntile = wid % 400; wid /= 400;
  int mtile = wid % 8;   wid /= 8;
  int b     = wid % 2;
  int plane = wid / 2;

  int rs, cs, po;
  const float* wconv; const float* bconv;
  if (plane == 0)      { rs = S;  cs = 1;  po = 40*SS; wconv = w_ax;  bconv = b_ax;  } // axial:   x[b,c,40,y,x]
  else if (plane == 1) { rs = 1;  cs = SS; po = 40*S;  wconv = w_cor; bconv = b_cor; } // coronal: x[b,c,col,40,row]
  else                 { rs = SS; cs = S;  po = 40;    wconv = w_sag; bconv = b_sag; } // sagittal:x[b,c,row,col,40]

  int row  = ntile / 5;
  int col0 = (ntile % 5) * 16;
  int mn   = lane & 15;          // M for A / N for B (shared)
  int kb2  = (lane >> 4) * 2;    // K-pair within 4-wide K block
  int oc0  = mtile * 16;
  int col  = col0 + mn;

  const float* xb     = xin + (long)b * CCH * SSS + po;
  const float* wbase0 = wconv + (long)(oc0 + mn) * (CCH * 9);

  v8f acc = {};
  for (int kr = 0; kr < 3; ++kr) {
    int rr = row + kr - 1;
    bool rok = (unsigned)rr < (unsigned)S;
    int rrc = rr < 0 ? 0 : (rr > S-1 ? S-1 : rr);      // clamped (always valid)
    for (int kc = 0; kc < 3; ++kc) {
      int cc = col + kc - 1;
      bool ok = rok && ((unsigned)cc < (unsigned)S);
      int ccc = cc < 0 ? 0 : (cc > S-1 ? S-1 : cc);
      const float* ip = xb + (long)rrc * rs + (long)ccc * cs;
      const float* wp = wbase0 + kr * 3 + kc;
      #pragma unroll 4
      for (int c0 = 0; c0 < CCH; c0 += 4) {
        int k0 = c0 + kb2;
        float b0 = ip[(long)k0 * SSS];                 // unconditional loads
        float b1 = ip[(long)(k0 + 1) * SSS];
        v2f a, bb;
        a.x  = wp[(long)k0 * 9];
        a.y  = wp[(long)(k0 + 1) * 9];
        bb.x = ok ? b0 : 0.0f;                         // v_cndmask, no branch
        bb.y = ok ? b1 : 0.0f;
        acc = __builtin_amdgcn_wmma_f32_16x16x4_f32(false, a, false, bb,
                                                    (short)0, acc, false, false);
      }
    }
  }
  // D layout: VGPR i -> M = i (+8 for lanes 16..31), N = lane&15
  float* out = conv_out + ((long)(plane * B_ + b) * CCH) * SS;
  int mhalf = (lane >> 4) * 8;
  #pragma unroll
  for (int i = 0; i < 8; ++i) {
    int oc = oc0 + mhalf + i;
    out[(long)oc * SS + row * S + col0 + mn] = acc[i] + bconv[oc];
  }
}

// ---------------------------------------------------------------------------
// Kernel 2: fa/fc/fs = W_fuse-slice (128x128) @ plane (128x6400), WMMA f32.
// ---------------------------------------------------------------------------
__global__ __launch_bounds__(256)
void fuse_gemm(const float* __restrict__ w_fuse, const float* __restrict__ conv_out,
               float* __restrict__ f_out) {
  int wid  = blockIdx.x * 8 + (threadIdx.x >> 5);
  int lane = threadIdx.x & 31;
  int ntile = wid % 400; wid /= 400;
  int mtile = wid % 8;   wid /= 8;
  int b     = wid % 2;
  int plane = wid / 2;

  int p0  = ntile * 16;
  int mn  = lane & 15;
  int kb2 = (lane >> 4) * 2;
  int oc0 = mtile * 16;

  const float* src  = conv_out + ((long)(plane * B_ + b) * CCH) * SS + p0 + mn;
  const float* wrow = w_fuse + (long)(oc0 + mn) * (3 * CCH) + plane * CCH;

  v8f acc = {};
  #pragma unroll 4
  for (int c0 = 0; c0 < CCH; c0 += 4) {
    int k0 = c0 + kb2;
    v2f a, bb;
    a.x  = wrow[k0];
    a.y  = wrow[k0 + 1];
    bb.x = src[(long)k0 * SS];
    bb.y = src[(long)(k0 + 1) * SS];
    acc = __builtin_amdgcn_wmma_f32_16x16x4_f32(false, a, false, bb,
                                                (short)0, acc, false, false);
  }
  float* out = f_out + ((long)(plane * B_ + b) * CCH) * SS;
  int mhalf = (lane >> 4) * 8;
  #pragma unroll
  for (int i = 0; i < 8; ++i) {
    int oc = oc0 + mhalf + i;
    out[(long)oc * SS + p0 + mn] = acc[i];
  }
}

// ---------------------------------------------------------------------------
// Kernel 3: per-plane marginals: S=sum, Q=sum of squares, Row[80], Col[80].
// One block per (plane,b,channel).
// ---------------------------------------------------------------------------
__global__ __launch_bounds__(128)
void plane_stats(const float* __restrict__ f_out, float* __restrict__ stats1) {
  __shared__ float redS[128];
  __shared__ float redQ[128];
  int pbo = blockIdx.x;
  const float* f = f_out + (long)pbo * SS;
  float* st = stats1 + (long)pbo * STAT_STRIDE;
  int t = threadIdx.x;
  float rsum = 0.f, q = 0.f;
  if (t < S) {
    const float* rp = f + t * S;
    for (int x2 = 0; x2 < S; ++x2) { float v = rp[x2]; rsum += v; q += v * v; }
    st[2 + t] = rsum;
  }
  redS[t] = (t < S) ? rsum : 0.f;
  redQ[t] = (t < S) ? q    : 0.f;
  __syncthreads();
  for (int off = 64; off > 0; off >>= 1) {
    if (t < off) { redS[t] += redS[t + off]; redQ[t] += redQ[t + off]; }
    __syncthreads();
  }
  if (t == 0) { st[0] = redS[0]; st[1] = redQ[0]; }
  if (t < S) {
    float csum = 0.f;
    for (int y = 0; y < S; ++y) csum += f[y * S + t];
    st[82 + t] = csum;
  }
}

// ---------------------------------------------------------------------------
// Kernel 4: analytic GroupNorm statistics from plane marginals.
// out = fa[y,x]+fc[z,x]+fs[z,y]+bf  =>  sums/cross-terms from marginals.
// One block per (b, group).
// ---------------------------------------------------------------------------
__global__ __launch_bounds__(128)
void group_stats(const float* __restrict__ stats1, const float* __restrict__ b_fuse,
                 float* __restrict__ stats2) {
  __shared__ float red[128];
  __shared__ float gacc[2];
  int bg = blockIdx.x;
  int b = bg >> 3, g = bg & 7;
  int t = threadIdx.x;
  if (t == 0) { gacc[0] = 0.f; gacc[1] = 0.f; }
  __syncthreads();
  for (int o = 0; o < 16; ++o) {
    int oc = g * 16 + o;
    const float* stA  = stats1 + (long)((0 * B_ + b) * CCH + oc) * STAT_STRIDE;
    const float* stC  = stats1 + (long)((1 * B_ + b) * CCH + oc) * STAT_STRIDE;
    const float* stSg = stats1 + (long)((2 * B_ + b) * CCH + oc) * STAT_STRIDE;
    float part = 0.f;
    if (t < S) {
      part = stA[82 + t] * stC[82 + t]     // sum_x ColA[x]*ColC[x]   (fa.fc)
           + stA[2 + t]  * stSg[82 + t]    // sum_y RowA[y]*ColS[y]   (fa.fs)
           + stC[2 + t]  * stSg[2 + t];    // sum_z RowC[z]*RowS[z]   (fc.fs)
    }
    red[t] = part;
    __syncthreads();
    for (int off = 64; off > 0; off >>= 1) {
      if (t < off) red[t] += red[t + off];
      __syncthreads();
    }
    if (t == 0) {
      float T  = red[0];
      float Sa = stA[0],  Qa = stA[1];
      float Sc = stC[0],  Qc = stC[1];
      float Ss = stSg[0], Qs = stSg[1];
      float bf = b_fuse[oc];
      float Ssum = Sa + Sc + Ss;
      float sum_o = 80.f * Ssum + 512000.f * bf;
      float ss_o  = 80.f * (Qa + Qc + Qs) + 2.f * T
                  + 2.f * bf * 80.f * Ssum + 512000.f * bf * bf;
      gacc[0] += sum_o; gacc[1] += ss_o;
    }
    __syncthreads();
  }
  if (t == 0) {
    const float Ng = 16.f * 512000.f;
    float mean = gacc[0] / Ng;
    float var  = gacc[1] / Ng - mean * mean;
    stats2[bg * 2]     = mean;
    stats2[bg * 2 + 1] = rsqrtf(var + 1e-5f);
  }
}

// ---------------------------------------------------------------------------
// Kernel 5: streaming pass: out = ((fa+fc+fs+bf)-mean)*inv*gn_w + gn_b + x.
// v4f (true clang vector) along x; x/out use non-temporal hints so the big
// streams don't evict the L2-resident fa/fc/fs planes.
// ---------------------------------------------------------------------------
__global__ __launch_bounds__(256)
void fuse_norm_add(const float* __restrict__ xin,
                   const float* __restrict__ f_out,
                   const float* __restrict__ stats2,
                   const float* __restrict__ b_fuse,
                   const float* __restrict__ gn_w, const float* __restrict__ gn_b,
                   float* __restrict__ outp) {
  long i4 = (long)blockIdx.x * blockDim.x + threadIdx.x;   // 32,768,000 total
  int xx4 = (int)(i4 % 20); long r = i4 / 20;
  int y = (int)(r % 80); r /= 80;
  int z = (int)(r % 80); r /= 80;
  int o = (int)(r % 128);
  int b = (int)(r / 128);

  const float* fa = f_out + ((long)(0 * B_ + b) * CCH + o) * SS;
  const float* fc = f_out + ((long)(1 * B_ + b) * CCH + o) * SS;
  const float* fs = f_out + ((long)(2 * B_ + b) * CCH + o) * SS;

  v4f va = ((const v4f*)(fa + y * S))[xx4];
  v4f vc = ((const v4f*)(fc + z * S))[xx4];
  float vs = fs[z * S + y];

  int g = o >> 4;
  float mean = stats2[(b * 8 + g) * 2];
  float inv  = stats2[(b * 8 + g) * 2 + 1];
  float w   = gn_w[o];
  float bb2 = gn_b[o];
  float base = vs + b_fuse[o] - mean;

  v4f xv = __builtin_nontemporal_load(((const v4f*)xin) + i4);
  v4f ov;
  ov.x = (va.x + vc.x + base) * inv * w + bb2 + xv.x;
  ov.y = (va.y + vc.y + base) * inv * w + bb2 + xv.y;
  ov.z = (va.z + vc.z + base) * inv * w + bb2 + xv.z;
  ov.w = (va.w + vc.w + base) * inv * w + bb2 + xv.w;
  __builtin_nontemporal_store(ov, ((v4f*)outp) + i4);
}

// ---------------------------------------------------------------------------
extern "C" void kernel_launch(void* const* d_in, const int* in_sizes, int n_in,
                              void* d_out, int out_size, void* d_ws, size_t ws_size,
                              hipStream_t stream) {
  const float* x      = (const float*)d_in[0];
  const float* w_ax   = (const float*)d_in[1];
  const float* b_ax   = (const float*)d_in[2];
  const float* w_cor  = (const float*)d_in[3];
  const float* b_cor  = (const float*)d_in[4];
  const float* w_sag  = (const float*)d_in[5];
  const float* b_sag  = (const float*)d_in[6];
  const float* w_fuse = (const float*)d_in[7];
  const float* b_fuse = (const float*)d_in[8];
  const float* gn_w   = (const float*)d_in[9];
  const float* gn_b   = (const float*)d_in[10];
  float* out = (float*)d_out;
  float* ws  = (float*)d_ws;

  float* conv_out = ws;
  float* f_out    = ws + FOUT_OFF;
  float* stats1   = ws + STAT1_OFF;
  float* stats2   = ws + STAT2_OFF;

  // 3 planes * 2 batch * 8 M-tiles * 400 N-tiles = 19200 waves = 2400 blocks
  conv_planes<<<2400, 256, 0, stream>>>(x, w_ax, b_ax, w_cor, b_cor, w_sag, b_sag, conv_out);
  fuse_gemm  <<<2400, 256, 0, stream>>>(w_fuse, conv_out, f_out);
  plane_stats<<<3 * B_ * CCH, 128, 0, stream>>>(f_out, stats1);
  group_stats<<<16, 128, 0, stream>>>(stats1, b_fuse, stats2);
  fuse_norm_add<<<128000, 256, 0, stream>>>(x, f_out, stats2, b_fuse, gn_w, gn_b, out);
}